// PointFeaturesAbstraction_34059090657805
// MI455X (gfx1250) — compile-verified
//
#include <hip/hip_runtime.h>
#include <hip/hip_bf16.h>
#include <math.h>

#define B_    2
#define N_    32768
#define K_    2048
#define NS_   16
#define CBEV_ 256
#define FEAT_ 288
#define COUT_ 128
#define M_    (B_ * K_)   // 4096

typedef float v2f __attribute__((ext_vector_type(2)));
typedef float v8f __attribute__((ext_vector_type(8)));

#if defined(__has_builtin)
#if __has_builtin(__builtin_amdgcn_ballot_w32)
#define BALLOT32(p) __builtin_amdgcn_ballot_w32(p)
#else
#define BALLOT32(p) ((unsigned)__ballot(p))
#endif
#else
#define BALLOT32(p) ((unsigned)__ballot(p))
#endif

__device__ __forceinline__ float relu_(float x) { return x > 0.0f ? x : 0.0f; }

// ---------------------------------------------------------------------------
// 1) Farthest point sampling: one 1024-thread workgroup per batch.
//    Each thread owns 32 points in registers (coords + running min-dist).
// ---------------------------------------------------------------------------
__global__ __launch_bounds__(1024) void fps_kernel(const float4* __restrict__ pts,
                                                   float* __restrict__ kp)
{
    const int b   = blockIdx.x;
    const int tid = threadIdx.x;
    const long base = (long)b * N_;

    float px[32], py[32], pz[32], dd[32];
#pragma unroll
    for (int j = 0; j < 32; ++j) {
        float4 p = pts[base + tid + (j << 10)];
        px[j] = p.x; py[j] = p.y; pz[j] = p.z; dd[j] = 1e10f;
    }

    __shared__ float swv[32];
    __shared__ int   swi[32];
    __shared__ int   s_last;

    if (tid == 0) {
        float4 q = pts[base];
        int o = (b * K_) * 4;
        kp[o] = q.x; kp[o + 1] = q.y; kp[o + 2] = q.z; kp[o + 3] = 0.0f;
    }

    int last = 0;
    const int wv = tid >> 5, lane = tid & 31;

    for (int t = 1; t < K_; ++t) {
        float4 lp = pts[base + last];   // uniform address, L2-hot broadcast
        float best = -1.0f;
        int   bi   = 0;
#pragma unroll
        for (int j = 0; j < 32; ++j) {
            float dx = px[j] - lp.x, dy = py[j] - lp.y, dz = pz[j] - lp.z;
            float d2 = dx * dx + dy * dy + dz * dz;
            float nd = fminf(dd[j], d2);
            dd[j] = nd;
            if (nd > best) { best = nd; bi = tid + (j << 10); }
        }
        // wave32 argmax reduce (first-occurrence tie break = smallest index)
#pragma unroll
        for (int off = 16; off >= 1; off >>= 1) {
            float ov = __shfl_down(best, off);
            int   oi = __shfl_down(bi, off);
            if (ov > best || (ov == best && oi < bi)) { best = ov; bi = oi; }
        }
        if (lane == 0) { swv[wv] = best; swi[wv] = bi; }
        __syncthreads();
        if (wv == 0) {
            float v  = swv[lane];
            int   i2 = swi[lane];
#pragma unroll
            for (int off = 16; off >= 1; off >>= 1) {
                float ov = __shfl_down(v, off);
                int   oi = __shfl_down(i2, off);
                if (ov > v || (ov == v && oi < i2)) { v = ov; i2 = oi; }
            }
            if (lane == 0) s_last = i2;
        }
        __syncthreads();
        last = s_last;
        if (tid == 0) {
            float4 q = pts[base + last];
            int o = (b * K_ + t) * 4;
            kp[o] = q.x; kp[o + 1] = q.y; kp[o + 2] = q.z; kp[o + 3] = 0.0f;
        }
    }
}

// ---------------------------------------------------------------------------
// 2) BEV bilinear interpolation: block = one keypoint, lane = channel.
// ---------------------------------------------------------------------------
__global__ __launch_bounds__(256) void bev_kernel(const float* __restrict__ bev,
                                                  const float* __restrict__ kp,
                                                  const int* __restrict__ stridep,
                                                  float* __restrict__ feat)
{
    const int k = blockIdx.x, b = blockIdx.y, c = threadIdx.x;
    const int gk = b * K_ + k;
    const float sf = (float)stridep[0];
    const float kx = kp[gk * 4 + 0], ky = kp[gk * 4 + 1];
    float xi = (kx - 0.0f) / 0.05f / sf;
    float yi = (ky + 40.0f) / 0.05f / sf;
    int fx = (int)floorf(xi), fy = (int)floorf(yi);
    int x0 = min(max(fx, 0), 175), x1 = min(max(fx + 1, 0), 175);
    int y0 = min(max(fy, 0), 199), y1 = min(max(fy + 1, 0), 199);
    const float* f = bev + (long)(b * CBEV_ + c) * (200 * 176);
    float Ia = f[y0 * 176 + x0], Ib = f[y1 * 176 + x0];
    float Ic = f[y0 * 176 + x1], Id = f[y1 * 176 + x1];
    float x0f = (float)x0, x1f = (float)x1, y0f = (float)y0, y1f = (float)y1;
    float wa = (x1f - xi) * (y1f - yi);
    float wb = (x1f - xi) * (yi - y0f);
    float wc = (xi - x0f) * (y1f - yi);
    float wd = (xi - x0f) * (yi - y0f);
    feat[(long)gk * FEAT_ + c] = Ia * wa + Ib * wb + Ic * wc + Id * wd;
}

// ---------------------------------------------------------------------------
// 3) Ball query for both radii: one wave per keypoint, ballot + prefix popcount
//    collects the first <=16 in-range indices in ascending order.
// ---------------------------------------------------------------------------
__global__ __launch_bounds__(256) void ballquery_kernel(const float4* __restrict__ pts,
                                                        const float* __restrict__ kp,
                                                        int* __restrict__ idx0,
                                                        int* __restrict__ idx1,
                                                        int* __restrict__ e0,
                                                        int* __restrict__ e1)
{
    const int wv = threadIdx.x >> 5, lane = threadIdx.x & 31;
    const int gk = blockIdx.x * 8 + wv;
    const int b  = gk / K_;
    const long base = (long)b * N_;
    const float kx = kp[gk * 4], ky = kp[gk * 4 + 1], kz = kp[gk * 4 + 2];

    __shared__ int l0[8][16];
    __shared__ int l1[8][16];
    int cnt0 = 0, cnt1 = 0;
    const unsigned lanemask = (1u << lane) - 1u;

    for (int bp = 0; bp < N_ && (cnt0 < 16 || cnt1 < 16); bp += 32) {
        int i = bp + lane;
        float4 p = pts[base + i];
        float dx = p.x - kx, dy = p.y - ky, dz = p.z - kz;
        float d2 = dx * dx + dy * dy + dz * dz;
        bool v0 = d2 < 0.16f;  // f32(0.4*0.4)
        bool v1 = d2 < 0.64f;  // f32(0.8*0.8)
        unsigned m0 = BALLOT32(v0);
        unsigned m1 = BALLOT32(v1);
        if (v0) { int s = cnt0 + __popc(m0 & lanemask); if (s < 16) l0[wv][s] = i; }
        cnt0 = min(16, cnt0 + __popc(m0));
        if (v1) { int s = cnt1 + __popc(m1 & lanemask); if (s < 16) l1[wv][s] = i; }
        cnt1 = min(16, cnt1 + __popc(m1));
    }

    if (lane < 16) {
        int v0 = (lane < cnt0) ? l0[wv][lane] : (cnt0 > 0 ? l0[wv][0] : 0);
        int v1 = (lane < cnt1) ? l1[wv][lane] : (cnt1 > 0 ? l1[wv][0] : 0);
        idx0[gk * 16 + lane] = v0;
        idx1[gk * 16 + lane] = v1;
    }
    if (lane == 0) { e0[gk] = (cnt0 == 0); e1[gk] = (cnt1 == 0); }
}

// ---------------------------------------------------------------------------
// 4) Group + layer1 (4->16) pre-activations + per-block BN partial sums.
//    Block = keypoint; thread = (sample s, channel c).
// ---------------------------------------------------------------------------
__global__ __launch_bounds__(256) void group_l1_kernel(const float4* __restrict__ pts,
                                                       const float* __restrict__ kp,
                                                       const int* __restrict__ idxl,
                                                       const int* __restrict__ el,
                                                       const float* __restrict__ w,
                                                       float* __restrict__ h1,
                                                       float* __restrict__ partial)
{
    const int gk = blockIdx.x;
    const int b  = gk / K_;
    const int t = threadIdx.x, s = t >> 4, c = t & 15;
    const int ni = idxl[gk * 16 + s];
    float4 p = pts[(long)b * N_ + ni];
    float f0 = p.x - kp[gk * 4 + 0];
    float f1 = p.y - kp[gk * 4 + 1];
    float f2 = p.z - kp[gk * 4 + 2];
    float f3 = p.w;
    if (el[gk]) { f0 = f1 = f2 = f3 = 0.0f; }
    float pre = f0 * w[c] + f1 * w[16 + c] + f2 * w[32 + c] + f3 * w[48 + c];
    h1[(long)gk * 256 + t] = pre;

    __shared__ float sh[256];
    sh[t] = pre;
    __syncthreads();
    if (t < 16) {
        float s1 = 0.0f, s2 = 0.0f;
#pragma unroll
        for (int i = 0; i < 16; ++i) { float v = sh[i * 16 + t]; s1 += v; s2 += v * v; }
        partial[gk * 32 + t]      = s1;
        partial[gk * 32 + 16 + t] = s2;
    }
}

// ---------------------------------------------------------------------------
// 5) Deterministic BN stats over 16 channels: stats[c] = rstd*g, stats[16+c] = b - m*rstd*g
// ---------------------------------------------------------------------------
__global__ __launch_bounds__(256) void bn_stats16_kernel(const float* __restrict__ partial,
                                                         int nblocks, float count,
                                                         const float* __restrict__ g,
                                                         const float* __restrict__ beta,
                                                         float* __restrict__ stats)
{
    const int t = threadIdx.x, c = t & 15, chunk = t >> 4;
    float s1 = 0.0f, s2 = 0.0f;
    for (int blk = chunk; blk < nblocks; blk += 16) {
        s1 += partial[blk * 32 + c];
        s2 += partial[blk * 32 + 16 + c];
    }
    __shared__ float a1[256], a2[256];
    a1[t] = s1; a2[t] = s2;
    __syncthreads();
    if (t < 16) {
        float S1 = 0.0f, S2 = 0.0f;
#pragma unroll
        for (int i = 0; i < 16; ++i) { S1 += a1[i * 16 + t]; S2 += a2[i * 16 + t]; }
        float m   = S1 / count;
        float var = S2 / count - m * m;
        float sc  = rsqrtf(var + 1e-5f) * g[t];
        stats[t]      = sc;
        stats[16 + t] = beta[t] - m * sc;
    }
}

// ---------------------------------------------------------------------------
// 6) BN1+ReLU then layer2 (16->16) pre-activations + BN partial sums.
// ---------------------------------------------------------------------------
__global__ __launch_bounds__(256) void l2_kernel(const float* __restrict__ h1,
                                                 const float* __restrict__ stats1,
                                                 const float* __restrict__ w,
                                                 float* __restrict__ h2,
                                                 float* __restrict__ partial)
{
    const int gk = blockIdx.x;
    const int t = threadIdx.x, s = t >> 4, c = t & 15;
    __shared__ float act[256];
    {
        float v = h1[(long)gk * 256 + t];
        act[t] = relu_(v * stats1[c] + stats1[16 + c]);
    }
    __syncthreads();
    float pre = 0.0f;
#pragma unroll
    for (int j = 0; j < 16; ++j) pre += act[s * 16 + j] * w[j * 16 + c];
    h2[(long)gk * 256 + t] = pre;

    __shared__ float sh[256];
    sh[t] = pre;
    __syncthreads();
    if (t < 16) {
        float s1 = 0.0f, s2 = 0.0f;
#pragma unroll
        for (int i = 0; i < 16; ++i) { float v = sh[i * 16 + t]; s1 += v; s2 += v * v; }
        partial[gk * 32 + t]      = s1;
        partial[gk * 32 + 16 + t] = s2;
    }
}

// ---------------------------------------------------------------------------
// 7) BN2+ReLU, max-pool over 16 samples, zero empties, scatter into feat.
// ---------------------------------------------------------------------------
__global__ __launch_bounds__(256) void pool_kernel(const float* __restrict__ h2,
                                                   const float* __restrict__ stats2,
                                                   const int* __restrict__ el,
                                                   float* __restrict__ feat, int coff)
{
    const int t = threadIdx.x, kl = t >> 4, c = t & 15;
    const int gk = blockIdx.x * 16 + kl;
    const float sc = stats2[c], sh = stats2[16 + c];
    float m = -1e30f;
#pragma unroll
    for (int s = 0; s < 16; ++s) {
        float v = h2[(long)gk * 256 + s * 16 + c];
        v = relu_(v * sc + sh);
        m = fmaxf(m, v);
    }
    if (el[gk]) m = 0.0f;
    feat[(long)gk * FEAT_ + coff + c] = m;
}

// ---------------------------------------------------------------------------
// 8) Final GEMM 4096x288 @ 288x128 in full f32 via V_WMMA_F32_16X16X4_F32.
//    Block = one 16-row M tile; wave = one 16-col N tile; 72 WMMAs per wave.
//    A layout (16x4 f32): lane L: M = L&15, K-pair = (L>>4)*2 (+0,+1 in vgpr0/1)
//    B layout (4x16 f32): lane L: N = L&15, same K-pair mapping.
//    D layout: vgpr r -> row r (+8 for upper lane half), col = lane&15.
// ---------------------------------------------------------------------------
__global__ __launch_bounds__(256) void gemm_wmma_kernel(const float* __restrict__ feat,
                                                        const float* __restrict__ wf,
                                                        float* __restrict__ pre)
{
    const int wv = threadIdx.x >> 5, lane = threadIdx.x & 31;
    const int m0 = blockIdx.x * 16;
    const int n0 = wv * 16;
    const int mrow = m0 + (lane & 15);
    const int ncol = n0 + (lane & 15);
    const int koff = (lane >> 4) * 2;

    v8f acc = {};
    const float* arow = feat + (long)mrow * FEAT_;
#pragma unroll 4
    for (int k0 = 0; k0 < FEAT_; k0 += 4) {
        v2f av = *reinterpret_cast<const v2f*>(arow + k0 + koff);
        v2f bv;
        bv.x = wf[(k0 + koff) * COUT_ + ncol];
        bv.y = wf[(k0 + koff + 1) * COUT_ + ncol];
        acc = __builtin_amdgcn_wmma_f32_16x16x4_f32(false, av, false, bv,
                                                    (short)0, acc, false, false);
    }
    const int rbase = m0 + ((lane >> 4) << 3);
#pragma unroll
    for (int r = 0; r < 8; ++r) {
        pre[(long)(rbase + r) * COUT_ + ncol] = acc[r];
    }
}

// ---------------------------------------------------------------------------
// 9) Final BN stats over axis 0 (4096 rows) per channel, deterministic.
// ---------------------------------------------------------------------------
__global__ __launch_bounds__(256) void final_stats_kernel(const float* __restrict__ pre,
                                                          const float* __restrict__ g,
                                                          const float* __restrict__ beta,
                                                          float* __restrict__ fstats)
{
    const int ch = blockIdx.x, t = threadIdx.x;
    float s1 = 0.0f, s2 = 0.0f;
    for (int r = t; r < M_; r += 256) {
        float v = pre[(long)r * COUT_ + ch];
        s1 += v; s2 += v * v;
    }
    __shared__ float r1[256], r2[256];
    r1[t] = s1; r2[t] = s2;
    __syncthreads();
    for (int o = 128; o > 0; o >>= 1) {
        if (t < o) { r1[t] += r1[t + o]; r2[t] += r2[t + o]; }
        __syncthreads();
    }
    if (t == 0) {
        float m   = r1[0] / (float)M_;
        float var = r2[0] / (float)M_ - m * m;
        float sc  = rsqrtf(var + 1e-5f) * g[ch];
        fstats[ch]         = sc;
        fstats[COUT_ + ch] = beta[ch] - m * sc;
    }
}

// ---------------------------------------------------------------------------
// 10) Apply BN + ReLU -> output.
// ---------------------------------------------------------------------------
__global__ __launch_bounds__(256) void final_apply_kernel(const float* __restrict__ pre,
                                                          const float* __restrict__ fstats,
                                                          float* __restrict__ out)
{
    const int i = blockIdx.x * 256 + threadIdx.x;
    const int c = i & (COUT_ - 1);
    out[i] = relu_(pre[i] * fstats[c] + fstats[COUT_ + c]);
}

// ---------------------------------------------------------------------------
extern "C" void kernel_launch(void* const* d_in, const int* in_sizes, int n_in,
                              void* d_out, int out_size, void* d_ws, size_t ws_size,
                              hipStream_t stream)
{
    const float4* pts  = (const float4*)d_in[0];
    const float* bev   = (const float*)d_in[1];
    const float* w0_0  = (const float*)d_in[2];
    const float* g0_0  = (const float*)d_in[3];
    const float* b0_0  = (const float*)d_in[4];
    const float* w0_1  = (const float*)d_in[5];
    const float* g0_1  = (const float*)d_in[6];
    const float* b0_1  = (const float*)d_in[7];
    const float* w1_0  = (const float*)d_in[8];
    const float* g1_0  = (const float*)d_in[9];
    const float* b1_0  = (const float*)d_in[10];
    const float* w1_1  = (const float*)d_in[11];
    const float* g1_1  = (const float*)d_in[12];
    const float* b1_1  = (const float*)d_in[13];
    const float* wf    = (const float*)d_in[14];
    const float* gf    = (const float*)d_in[15];
    const float* bf    = (const float*)d_in[16];
    const int* bstride = (const int*)d_in[17];
    (void)in_sizes; (void)n_in; (void)out_size; (void)ws_size;

    char* ws = (char*)d_ws;
    size_t off = 0;
    auto take = [&](size_t bytes) -> char* {
        char* p = ws + off;
        off = (off + bytes + 255) & ~(size_t)255;
        return p;
    };
    float* kp      = (float*)take((size_t)B_ * K_ * 4 * sizeof(float));
    int*   idx0    = (int*)  take((size_t)B_ * K_ * 16 * sizeof(int));
    int*   idx1    = (int*)  take((size_t)B_ * K_ * 16 * sizeof(int));
    int*   e0      = (int*)  take((size_t)B_ * K_ * sizeof(int));
    int*   e1      = (int*)  take((size_t)B_ * K_ * sizeof(int));
    float* h1      = (float*)take((size_t)B_ * K_ * 256 * sizeof(float));
    float* h2      = (float*)take((size_t)B_ * K_ * 256 * sizeof(float));
    float* partial = (float*)take((size_t)B_ * K_ * 32 * sizeof(float));
    float* stats1  = (float*)take(32 * sizeof(float));
    float* stats2  = (float*)take(32 * sizeof(float));
    float* feat    = (float*)take((size_t)M_ * FEAT_ * sizeof(float));
    float* gpre    = (float*)take((size_t)M_ * COUT_ * sizeof(float));
    float* fstats  = (float*)take((size_t)2 * COUT_ * sizeof(float));

    const float cnt = (float)(M_ * NS_);

    fps_kernel<<<B_, 1024, 0, stream>>>(pts, kp);
    bev_kernel<<<dim3(K_, B_), 256, 0, stream>>>(bev, kp, bstride, feat);
    ballquery_kernel<<<M_ / 8, 256, 0, stream>>>(pts, kp, idx0, idx1, e0, e1);

    // radius 0 pipeline
    group_l1_kernel<<<M_, 256, 0, stream>>>(pts, kp, idx0, e0, w0_0, h1, partial);
    bn_stats16_kernel<<<1, 256, 0, stream>>>(partial, M_, cnt, g0_0, b0_0, stats1);
    l2_kernel<<<M_, 256, 0, stream>>>(h1, stats1, w0_1, h2, partial);
    bn_stats16_kernel<<<1, 256, 0, stream>>>(partial, M_, cnt, g0_1, b0_1, stats2);
    pool_kernel<<<M_ / 16, 256, 0, stream>>>(h2, stats2, e0, feat, CBEV_);

    // radius 1 pipeline (reuses h1/h2/partial/stats buffers; stream-serial)
    group_l1_kernel<<<M_, 256, 0, stream>>>(pts, kp, idx1, e1, w1_0, h1, partial);
    bn_stats16_kernel<<<1, 256, 0, stream>>>(partial, M_, cnt, g1_0, b1_0, stats1);
    l2_kernel<<<M_, 256, 0, stream>>>(h1, stats1, w1_1, h2, partial);
    bn_stats16_kernel<<<1, 256, 0, stream>>>(partial, M_, cnt, g1_1, b1_1, stats2);
    pool_kernel<<<M_ / 16, 256, 0, stream>>>(h2, stats2, e1, feat, CBEV_ + 16);

    gemm_wmma_kernel<<<M_ / 16, 256, 0, stream>>>(feat, wf, gpre);
    final_stats_kernel<<<COUT_, 256, 0, stream>>>(gpre, gf, bf, fstats);
    final_apply_kernel<<<(M_ * COUT_) / 256, 256, 0, stream>>>(gpre, fstats, (float*)d_out);
}